// Attn_29815662969299
// MI455X (gfx1250) — compile-verified
//
#include <hip/hip_runtime.h>
#include <math.h>
#include <stdint.h>

// Attn(method='dot'): scores = enc[L,H] @ hidden[H]; softmax over L; out [1,L].
// L = 262144, H = 512, fp32. HBM-bound: 512MB streamed once (~22us @ 23.3TB/s).
// Phase 1: V_WMMA_F32_16X16X4_F32 (16 rows/wave, hidden chunk replicated across
// the 16 B-columns). Tiles are staged with GLOBAL_LOAD_ASYNC_TO_LDS_B128 into a
// wave-private double-buffered LDS slab (ASYNCcnt-tracked, no VGPR staging, no
// block barriers in the steady-state loop).

#define HIDDEN        512
#define MAX_LEN       262144
#define ROWS_PER_WAVE 16
#define WAVES         8
#define ROWS_PER_BLK  (ROWS_PER_WAVE * WAVES)   // 128
#define KCHUNK        32
#define NCHUNK        (HIDDEN / KCHUNK)         // 16
#define LDS_STRIDE    36                        // 32 + 4 pad: (36*m + c) % 64 distinct -> no bank conflicts
#define NPART         512

typedef __attribute__((ext_vector_type(2))) float v2f;
typedef __attribute__((ext_vector_type(8))) float v8f;

__global__ __launch_bounds__(256) void attn_scores_wmma(
    const float* __restrict__ enc, const float* __restrict__ hidden,
    float* __restrict__ scores)
{
    // [wave][double-buffer][16 x 36] + hidden copy: ~39KB static LDS
    __shared__ __align__(16) float ldsA[WAVES][2][ROWS_PER_WAVE * LDS_STRIDE];
    __shared__ __align__(16) float ldsH[HIDDEN];

    const int tid  = threadIdx.x;
    const int wave = tid >> 5;
    const int lane = tid & 31;
    const int h    = lane >> 4;   // half-wave: selects K pair {0,1} vs {2,3}
    const int l15  = lane & 15;   // A-matrix row within the 16-row tile

    // Stage the (tiny) hidden vector once per block.
    ldsH[tid]       = hidden[tid];
    ldsH[tid + 256] = hidden[tid + 256];
    __syncthreads();              // only block-wide barrier needed

    const int row_base = blockIdx.x * ROWS_PER_BLK + wave * ROWS_PER_WAVE;

    // Per-lane slice of the 16x32 chunk: lane covers row r0+4i, cols c0l..c0l+3.
    const int r0  = lane >> 3;          // 0..3
    const int c0l = (lane * 4) & 31;    // 0,4,...,28

    // LDS byte offset of this wave's slab (LDS aperture: low 32 bits of the
    // generic address are the workgroup-relative LDS byte address).
    const unsigned ldsbase = (unsigned)(uintptr_t)(const void*)&ldsA[wave][0][0];

    auto issue_chunk = [&](int kc) {
        const int      b  = kc & 1;
        const int      k0 = kc * KCHUNK;
        #pragma unroll
        for (int i = 0; i < 4; ++i) {
            const int      r    = 4 * i + r0;
            const unsigned loff = ldsbase +
                (unsigned)((b * ROWS_PER_WAVE * LDS_STRIDE + r * LDS_STRIDE + c0l) * 4);
            const float*   g    = &enc[(size_t)(row_base + r) * HIDDEN + k0 + c0l];
            // 16B/lane, fully coalesced (each 8-lane group covers one 128B row run)
            asm volatile("global_load_async_to_lds_b128 %0, %1, off"
                         :: "v"(loff), "v"(g)
                         : "memory");
        }
    };

    issue_chunk(0);

    v8f acc0 = {};
    v8f acc1 = {};

    for (int kc = 0; kc < NCHUNK; ++kc) {
        if (kc + 1 < NCHUNK) {
            issue_chunk(kc + 1);  // prefetch next chunk into the other buffer
            // async loads complete in order: <=4 outstanding means the 4 for
            // chunk kc (issued one iteration ago) have landed in LDS.
            asm volatile("s_wait_asynccnt 0x4" ::: "memory");
        } else {
            asm volatile("s_wait_asynccnt 0x0" ::: "memory");
        }

        const float* bufA = &ldsA[wave][kc & 1][0];
        const int    k0   = kc * KCHUNK;

        // 8 WMMAs per chunk (K=4 each), 2 accumulators to break the RAW chain.
        #pragma unroll
        for (int kk = 0; kk < 8; kk += 2) {
            const int cA = kk * 4 + 2 * h;
            v2f a0 = *(const v2f*)&bufA[l15 * LDS_STRIDE + cA];
            v2f b0 = *(const v2f*)&ldsH[k0 + cA];
            acc0 = __builtin_amdgcn_wmma_f32_16x16x4_f32(
                false, a0, false, b0, (short)0, acc0, false, false);

            const int cB = (kk + 1) * 4 + 2 * h;
            v2f a1 = *(const v2f*)&bufA[l15 * LDS_STRIDE + cB];
            v2f b1 = *(const v2f*)&ldsH[k0 + cB];
            acc1 = __builtin_amdgcn_wmma_f32_16x16x4_f32(
                false, a1, false, b1, (short)0, acc1, false, false);
        }
    }

    // C layout: VGPR r, lane l -> row r + 8*(l>=16), col l&15. Column 0 holds
    // the scores; lanes 0 and 16 each write 8 rows.
    if (l15 == 0) {
        #pragma unroll
        for (int r = 0; r < 8; ++r) {
            scores[row_base + 8 * h + r] = acc0[r] + acc1[r];
        }
    }
}

// ---- softmax epilogue: online (max, sumexp) reduction over 1MB of scores ----

__device__ __forceinline__ void merge_ms(float& m, float& s, float m2, float s2)
{
    const float mn = fmaxf(m, m2);
    s = s * expf(m - mn) + s2 * expf(m2 - mn);
    m = mn;
}

__global__ __launch_bounds__(256) void attn_partial(
    const float* __restrict__ scores, float2* __restrict__ part)
{
    __shared__ float sm[256];
    __shared__ float ss[256];
    const int tid  = threadIdx.x;
    const int base = blockIdx.x * (MAX_LEN / NPART);

    float m = -INFINITY, s = 0.0f;
    for (int i = tid; i < MAX_LEN / NPART; i += 256) {
        const float x  = scores[base + i];
        const float mn = fmaxf(m, x);
        s = s * expf(m - mn) + expf(x - mn);
        m = mn;
    }
    sm[tid] = m; ss[tid] = s;
    __syncthreads();
    for (int off = 128; off > 0; off >>= 1) {
        if (tid < off) {
            float mm = sm[tid], sv = ss[tid];
            merge_ms(mm, sv, sm[tid + off], ss[tid + off]);
            sm[tid] = mm; ss[tid] = sv;
        }
        __syncthreads();
    }
    if (tid == 0) part[blockIdx.x] = make_float2(sm[0], ss[0]);
}

__global__ __launch_bounds__(256) void attn_final(
    const float2* __restrict__ part, float2* __restrict__ ms_out)
{
    __shared__ float sm[256];
    __shared__ float ss[256];
    const int tid = threadIdx.x;

    float2 p0 = part[tid];
    float2 p1 = part[tid + 256];
    float m = p0.x, s = p0.y;
    merge_ms(m, s, p1.x, p1.y);
    sm[tid] = m; ss[tid] = s;
    __syncthreads();
    for (int off = 128; off > 0; off >>= 1) {
        if (tid < off) {
            float mm = sm[tid], sv = ss[tid];
            merge_ms(mm, sv, sm[tid + off], ss[tid + off]);
            sm[tid] = mm; ss[tid] = sv;
        }
        __syncthreads();
    }
    if (tid == 0) ms_out[0] = make_float2(sm[0], ss[0]);
}

__global__ __launch_bounds__(256) void attn_norm(
    const float* __restrict__ scores, const float2* __restrict__ ms,
    float* __restrict__ out)
{
    const int i = blockIdx.x * 256 + threadIdx.x;
    const float2 v = *ms;
    out[i] = expf(scores[i] - v.x) / v.y;
}

extern "C" void kernel_launch(void* const* d_in, const int* in_sizes, int n_in,
                              void* d_out, int out_size, void* d_ws, size_t ws_size,
                              hipStream_t stream)
{
    (void)in_sizes; (void)n_in; (void)out_size; (void)ws_size;

    const float* hidden = (const float*)d_in[0];   // [512]
    const float* enc    = (const float*)d_in[1];   // [262144, 512]
    float*       out    = (float*)d_out;           // [1, 262144] fp32

    // Workspace layout: scores (1MB) | NPART float2 partials | final float2
    float*  scores   = (float*)d_ws;
    float2* partials = (float2*)(scores + MAX_LEN);
    float2* finalMS  = partials + NPART;

    attn_scores_wmma<<<MAX_LEN / ROWS_PER_BLK, 256, 0, stream>>>(enc, hidden, scores);
    attn_partial<<<NPART, 256, 0, stream>>>(scores, partials);
    attn_final<<<1, 256, 0, stream>>>(partials, finalMS);
    attn_norm<<<MAX_LEN / 256, 256, 0, stream>>>(scores, finalMS, out);
}